// DynamicClusterEmbeddingModel_26886495273500
// MI455X (gfx1250) — compile-verified
//
#include <hip/hip_runtime.h>

#define EPSF 1e-10f
#define LDS_STRIDE 34   // even stride: keeps K-pair loads 8B-aligned, avoids systematic bank conflicts

typedef __attribute__((ext_vector_type(2))) float v2f;
typedef __attribute__((ext_vector_type(8))) float v8f;

// D = A(16x4) * B(4x16) + C(16x16), all f32, wave32.
// Per-lane layouts (from CDNA5 ISA 7.12.2):
//   A: lane l -> row (l&15), K = 2*(l>>4) in v0, +1 in v1
//   B: lane l -> col (l&15), K = 2*(l>>4) in v0, +1 in v1
//   C/D: vgpr v, lane l -> row v + 8*(l>>4), col (l&15)
__device__ __forceinline__ v8f wmma_f32_4(v2f a, v2f b, v8f c) {
    return __builtin_amdgcn_wmma_f32_16x16x4_f32(
        /*neg_a=*/false, a, /*neg_b=*/false, b,
        /*c_mod=*/(short)0, c, /*reuse_a=*/false, /*reuse_b=*/false);
}

__global__ __launch_bounds__(256) void DynamicClusterEmbeddingModel_kernel(
    const long long* __restrict__ cluster_ids,
    const float* __restrict__ emb,
    const float* __restrict__ w1, const float* __restrict__ b1,
    const float* __restrict__ w2, const float* __restrict__ b2,
    const float* __restrict__ w3, const float* __restrict__ b3,
    const float* __restrict__ uniform,
    float* __restrict__ out,
    int N, int K)
{
    __shared__ float h1[64 * LDS_STRIDE];
    __shared__ float h2[64 * LDS_STRIDE];
    __shared__ float probs[64];

    const int tid = threadIdx.x;

    // ---------------- Phase 1: wave 0 computes the 2 hidden layers with f32 WMMA.
    // E (Kx32, padded to 64x32) @ W (32x32): 4 M-tiles x 2 N-tiles x 8 k-steps of K=4.
    if (tid < 32) {
        const int lane  = tid;
        const int mrow  = lane & 15;   // row (A) / col (B,D) within tile
        const int khalf = lane >> 4;   // 0 -> K{0,1}, 1 -> K{2,3} of each 4-chunk

        // ---- Layer 1: h1 = relu(emb @ w1 + b1)
        #pragma unroll
        for (int Mt = 0; Mt < 4; ++Mt) {
            v8f acc0 = {}; v8f acc1 = {};
            const int row_g  = Mt * 16 + mrow;
            const bool valid = row_g < K;
            const int row_c  = valid ? row_g : (K - 1);   // clamp: load real data, zero via cndmask
            #pragma unroll
            for (int kt = 0; kt < 8; ++kt) {
                const int kb = kt * 4 + 2 * khalf;
                // unconditional 8B load + branch-free zeroing (v_cndmask, no exec games)
                v2f a = *(const v2f*)(emb + row_c * 32 + kb);
                a.x = valid ? a.x : 0.0f;
                a.y = valid ? a.y : 0.0f;
                v2f bA, bB;
                bA.x = w1[(kb    ) * 32 +      mrow];
                bA.y = w1[(kb + 1) * 32 +      mrow];
                bB.x = w1[(kb    ) * 32 + 16 + mrow];
                bB.y = w1[(kb + 1) * 32 + 16 + mrow];
                acc0 = wmma_f32_4(a, bA, acc0);
                acc1 = wmma_f32_4(a, bB, acc1);
            }
            const float bias0 = b1[mrow];
            const float bias1 = b1[16 + mrow];
            #pragma unroll
            for (int v = 0; v < 8; ++v) {
                const int r = Mt * 16 + v + 8 * khalf;
                h1[r * LDS_STRIDE +      mrow] = fmaxf(acc0[v] + bias0, 0.0f);
                h1[r * LDS_STRIDE + 16 + mrow] = fmaxf(acc1[v] + bias1, 0.0f);
            }
        }
        asm volatile("s_wait_dscnt 0x0" ::: "memory");   // same-wave LDS RAW fence

        // ---- Layer 2: h2 = relu(h1 @ w2 + b2)
        #pragma unroll
        for (int Mt = 0; Mt < 4; ++Mt) {
            v8f acc0 = {}; v8f acc1 = {};
            #pragma unroll
            for (int kt = 0; kt < 8; ++kt) {
                const int kb = kt * 4 + 2 * khalf;
                // (row*34 + kb) is even -> 8B-aligned ds_load_b64
                v2f a = *(const v2f*)&h1[(Mt * 16 + mrow) * LDS_STRIDE + kb];
                v2f bA, bB;
                bA.x = w2[(kb    ) * 32 +      mrow];
                bA.y = w2[(kb + 1) * 32 +      mrow];
                bB.x = w2[(kb    ) * 32 + 16 + mrow];
                bB.y = w2[(kb + 1) * 32 + 16 + mrow];
                acc0 = wmma_f32_4(a, bA, acc0);
                acc1 = wmma_f32_4(a, bB, acc1);
            }
            const float bias0 = b2[mrow];
            const float bias1 = b2[16 + mrow];
            #pragma unroll
            for (int v = 0; v < 8; ++v) {
                const int r = Mt * 16 + v + 8 * khalf;
                h2[r * LDS_STRIDE +      mrow] = fmaxf(acc0[v] + bias0, 0.0f);
                h2[r * LDS_STRIDE + 16 + mrow] = fmaxf(acc1[v] + bias1, 0.0f);
            }
        }
    }
    __syncthreads();

    // ---------------- Phase 2: logits, Gumbel-style noise, sigmoid (one thread per cluster)
    if (tid < K) {
        float acc = 0.0f;
        #pragma unroll
        for (int c = 0; c < 32; ++c)
            acc += h2[tid * LDS_STRIDE + c] * w3[c];
        const float logit = acc + b3[0];
        float u0 = fminf(fmaxf(uniform[tid],     EPSF), 1.0f - EPSF);
        float u1 = fminf(fmaxf(uniform[K + tid], EPSF), 1.0f - EPSF);
        const float noise = -logf(logf(u1) / logf(u0) + EPSF);
        const float x = logit + noise;
        probs[tid] = 1.0f / (1.0f + expf(-x));
    }
    __syncthreads();

    // ---------------- Phase 3: streaming gather, 4 elements/thread (2x16B loads, 1x16B store)
    const long long base = ((long long)blockIdx.x * blockDim.x + tid) * 4LL;
    if (base + 3 < (long long)N) {
        const int4* ids4 = (const int4*)cluster_ids;     // one int4 = two int64 ids (values < 50 -> low dwords)
        const int4 p0 = ids4[(size_t)(base >> 1)];
        const int4 p1 = ids4[(size_t)(base >> 1) + 1];
        float4 o;
        o.x = probs[p0.x];
        o.y = probs[p0.z];
        o.z = probs[p1.x];
        o.w = probs[p1.z];
        ((float4*)out)[(size_t)(base >> 2)] = o;
    } else {
        #pragma unroll
        for (int k = 0; k < 4; ++k) {
            const long long i = base + k;
            if (i < (long long)N)
                out[i] = probs[(int)cluster_ids[i]];
        }
    }
}

extern "C" void kernel_launch(void* const* d_in, const int* in_sizes, int n_in,
                              void* d_out, int out_size, void* d_ws, size_t ws_size,
                              hipStream_t stream) {
    // setup_inputs order:
    // 0: feats (unused by the reference output), 1: cluster_ids (int64),
    // 2: emb (K,32), 3: w1 (32,32), 4: b1 (32), 5: w2 (32,32), 6: b2 (32),
    // 7: w3 (32,1), 8: b3 (1), 9: uniform (2,K,1)
    const long long* ids = (const long long*)d_in[1];
    const float* emb = (const float*)d_in[2];
    const float* w1  = (const float*)d_in[3];
    const float* b1  = (const float*)d_in[4];
    const float* w2  = (const float*)d_in[5];
    const float* b2  = (const float*)d_in[6];
    const float* w3  = (const float*)d_in[7];
    const float* b3  = (const float*)d_in[8];
    const float* uni = (const float*)d_in[9];
    float* out = (float*)d_out;

    const int N = in_sizes[1];        // 2,000,000
    const int K = in_sizes[9] / 2;    // 50

    const int elems_per_block = 256 * 4;
    const int nblocks = (N + elems_per_block - 1) / elems_per_block;

    DynamicClusterEmbeddingModel_kernel<<<nblocks, 256, 0, stream>>>(
        ids, emb, w1, b1, w2, b2, w3, b3, uni, out, N, K);
}